// DeeperGCN_44555990729011
// MI455X (gfx1250) — compile-verified
//
#include <hip/hip_runtime.h>
#include <hip/hip_bf16.h>
#include <utility>
#include <stdint.h>

typedef __attribute__((ext_vector_type(2))) float v2f;
typedef __attribute__((ext_vector_type(8))) float v8f;
typedef __attribute__((ext_vector_type(4))) unsigned int u32x4;
typedef __attribute__((ext_vector_type(8))) int i32x8;
typedef __attribute__((ext_vector_type(4))) int i32x4;

// Problem constants (match reference)
constexpr int   Nn      = 50000;
constexpr int   Ee      = 400000;
constexpr int   Dd      = 64;
constexpr int   Gg      = 256;
constexpr int   Ll      = 4;
constexpr float MSG_EPS = 1e-7f;
constexpr float LN_EPS  = 1e-5f;

#if defined(__has_builtin)
#if __has_builtin(__builtin_amdgcn_tensor_load_to_lds) && \
    __has_builtin(__builtin_amdgcn_s_wait_tensorcnt)
#define HAVE_TDM 1
#endif
#endif
#ifndef HAVE_TDM
#define HAVE_TDM 0
#endif

// -------------------------------------------------------------------------
// Node init: h[n,:] = sum_f atom_emb[f, x[n,f], :] + vn_emb   (vn starts as
// a broadcast of vn_emb, so the initial h += vn[batch] is just += vn_emb)
// One wave32 per node; lane handles features (2*lane, 2*lane+1).
__global__ __launch_bounds__(256) void k_atom_encode(
    const int* __restrict__ x, const float* __restrict__ atom_emb,
    const float* __restrict__ vn_emb, float* __restrict__ h) {
  const int lane = threadIdx.x & 31;
  const int wid  = (blockIdx.x * blockDim.x + threadIdx.x) >> 5;
  const int nw   = (gridDim.x * blockDim.x) >> 5;
  for (int n = wid; n < Nn; n += nw) {
    float2 acc = ((const float2*)vn_emb)[lane];
#pragma unroll
    for (int f = 0; f < 9; ++f) {
      int idx = x[n * 9 + f];                 // vocab VA = 64
      float2 e = ((const float2*)(atom_emb + ((size_t)f * 64 + idx) * Dd))[lane];
      acc.x += e.x; acc.y += e.y;
    }
    ((float2*)(h + (size_t)n * Dd))[lane] = acc;
  }
}

__global__ void k_init_vn(const float* __restrict__ vn_emb, float* __restrict__ vn) {
  int i = blockIdx.x * blockDim.x + threadIdx.x;
  if (i < Gg * Dd) vn[i] = vn_emb[i & (Dd - 1)];
}

// -------------------------------------------------------------------------
// Edge pass 1: msg = relu(h[src]+bond_emb) + eps ; per-(dst,feat) max.
// Messages are strictly > 0, so int-bitwise atomicMax implements float max,
// and 0-init matches the reference's "no in-edge -> mmax := 0" semantics.
// Bond table (6 KB) and h (12.8 MB) are L2-resident -> recompute, never
// materialize the [E,64] message tensor (saves ~100 MB HBM per pass).
__global__ __launch_bounds__(256) void k_msg_max(
    const float* __restrict__ hx, const int* __restrict__ src,
    const int* __restrict__ dst, const int* __restrict__ eattr,
    const float* __restrict__ bond_emb, float* __restrict__ mmax) {
  const int lane = threadIdx.x & 31;
  const int wid  = (blockIdx.x * blockDim.x + threadIdx.x) >> 5;
  const int nw   = (gridDim.x * blockDim.x) >> 5;
  for (int e = wid; e < Ee; e += nw) {
    int en = e + nw;
    if (en < Ee) {                      // software prefetch of next edge stream
      __builtin_prefetch(&eattr[3 * en], 0, 0);
      __builtin_prefetch(&src[en], 0, 0);
    }
    int s = src[e], t = dst[e];
    int a0 = eattr[3 * e], a1 = eattr[3 * e + 1], a2 = eattr[3 * e + 2];
    float2 hv = ((const float2*)(hx + (size_t)s * Dd))[lane];
    float2 b0 = ((const float2*)(bond_emb + (size_t)(0 * 8 + a0) * Dd))[lane];
    float2 b1 = ((const float2*)(bond_emb + (size_t)(1 * 8 + a1) * Dd))[lane];
    float2 b2 = ((const float2*)(bond_emb + (size_t)(2 * 8 + a2) * Dd))[lane];
    float mx = fmaxf(hv.x + b0.x + b1.x + b2.x, 0.f) + MSG_EPS;
    float my = fmaxf(hv.y + b0.y + b1.y + b2.y, 0.f) + MSG_EPS;
    atomicMax((int*)&mmax[(size_t)t * Dd + 2 * lane],     __float_as_int(mx));
    atomicMax((int*)&mmax[(size_t)t * Dd + 2 * lane + 1], __float_as_int(my));
  }
}

// Edge pass 2: ex = exp(msg - mmax[dst]); denom += ex; num += ex*msg.
// (softmax-weighted sum = num/denom, so only two N x D accumulators needed)
__global__ __launch_bounds__(256) void k_msg_sum(
    const float* __restrict__ hx, const int* __restrict__ src,
    const int* __restrict__ dst, const int* __restrict__ eattr,
    const float* __restrict__ bond_emb, const float* __restrict__ mmax,
    float* __restrict__ num, float* __restrict__ den) {
  const int lane = threadIdx.x & 31;
  const int wid  = (blockIdx.x * blockDim.x + threadIdx.x) >> 5;
  const int nw   = (gridDim.x * blockDim.x) >> 5;
  for (int e = wid; e < Ee; e += nw) {
    int en = e + nw;
    if (en < Ee) {
      __builtin_prefetch(&eattr[3 * en], 0, 0);
      __builtin_prefetch(&src[en], 0, 0);
    }
    int s = src[e], t = dst[e];
    int a0 = eattr[3 * e], a1 = eattr[3 * e + 1], a2 = eattr[3 * e + 2];
    float2 hv = ((const float2*)(hx + (size_t)s * Dd))[lane];
    float2 b0 = ((const float2*)(bond_emb + (size_t)(0 * 8 + a0) * Dd))[lane];
    float2 b1 = ((const float2*)(bond_emb + (size_t)(1 * 8 + a1) * Dd))[lane];
    float2 b2 = ((const float2*)(bond_emb + (size_t)(2 * 8 + a2) * Dd))[lane];
    float mx = fmaxf(hv.x + b0.x + b1.x + b2.x, 0.f) + MSG_EPS;
    float my = fmaxf(hv.y + b0.y + b1.y + b2.y, 0.f) + MSG_EPS;
    float2 mm = ((const float2*)(mmax + (size_t)t * Dd))[lane];
    float ex = __expf(mx - mm.x);
    float ey = __expf(my - mm.y);
    float* dn = den + (size_t)t * Dd + 2 * lane;
    float* nm = num + (size_t)t * Dd + 2 * lane;
    atomicAdd(dn,     ex);
    atomicAdd(dn + 1, ey);
    atomicAdd(nm,     ex * mx);
    atomicAdd(nm + 1, ey * my);
  }
}

// num <- hx + num / max(den, 1e-16)   (GEMM input, in place)
__global__ void k_finalize(const float* __restrict__ hx, const float* __restrict__ den,
                           float* __restrict__ num) {
  int i = blockIdx.x * blockDim.x + threadIdx.x;
  const int tot = Nn * Dd;
  for (; i < tot; i += gridDim.x * blockDim.x)
    num[i] = hx[i] + num[i] / fmaxf(den[i], 1e-16f);
}

// -------------------------------------------------------------------------
// out[M,64] = in[M,64] @ W[64,64] + bias (+ res).  One wave per 16-row tile,
// W staged into LDS by the Tensor Data Mover (one 16 KB 2-D D# descriptor
// issued by wave 0, TENSORcnt-tracked), 16 K-steps x 4 N-tiles of
// V_WMMA_F32_16X16X4_F32 per tile.
// A 16x4 f32 layout (ISA 7.12.2): v0 = {lanes0-15: K=k0, lanes16-31: K=k0+2},
// v1 = {K=k0+1, K=k0+3}; B mirrored; C/D: vgpr i -> row i (+8 for hi lanes).
// M must be a multiple of 16 (50000 and 256 both are).
__global__ __launch_bounds__(256) void k_gemm64(
    const float* __restrict__ in, const float* __restrict__ W,
    const float* __restrict__ bias, const float* __restrict__ res,
    float* __restrict__ out, int M) {
  __shared__ float sW[64 * 64];
#if HAVE_TDM
  if (threadIdx.x < 32) {   // wave 0 issues one TDM descriptor for all 16 KB
    const unsigned long long ga = (unsigned long long)(uintptr_t)W;
    // Generic pointer to LDS: aperture check maps low 32 bits to LDS offset.
    const unsigned lds = (unsigned)(uintptr_t)(void*)&sW[0];
    u32x4 g0;
    g0[0] = 1u;                                   // count=1, user descriptor
    g0[1] = lds;                                  // lds_addr
    g0[2] = (unsigned)(ga & 0xFFFFFFFFu);         // global_addr[31:0]
    g0[3] = (unsigned)((ga >> 32) & 0x01FFFFFFu)  // global_addr[56:32]
          | (2u << 30);                           // type = 2 ("image")
    i32x8 g1;
    g1[0] = 0x00020000;         // wg_mask=0, data_size=2 (4 B), no extras
    g1[1] = (int)(64u << 16);   // tensor_dim0 = 64   (bits 79:48, lo16)
    g1[2] = (int)(64u << 16);   // tensor_dim1 = 64   (bits 111:80, lo16)
    g1[3] = (int)(64u << 16);   // tile_dim0   = 64   (bits 127:112)
    g1[4] = 64;                 // tile_dim1   = 64   (bits 143:128), tile_dim2=0
    g1[5] = 64;                 // tensor_dim0_stride = 64 (bits 207:160, lo32)
    g1[6] = 0;                  // stride hi / dim1_stride (unused for 2-D tile)
    g1[7] = 0;
    i32x4 gz4 = {};             // groups 2/3 unused (<=2-D tensor)
    i32x8 gz8 = {};             // trailing group (this toolchain's 6-arg form)
    __builtin_amdgcn_tensor_load_to_lds(g0, g1, gz4, gz4, gz8, 0);
    __builtin_amdgcn_s_wait_tensorcnt(0);
  }
  __syncthreads();
#else
  for (int i = threadIdx.x; i < 64 * 64; i += 256) sW[i] = W[i];
  __syncthreads();
#endif
  const int lane  = threadIdx.x & 31;
  const int wid   = threadIdx.x >> 5;
  const int lrow  = lane & 15;
  const int khalf = (lane >> 4) << 1;          // 0 or 2
  const int tiles = M >> 4;
  for (int t = blockIdx.x * 8 + wid; t < tiles; t += gridDim.x * 8) {
    const int row0 = t << 4;
    const float* arow = in + (size_t)(row0 + lrow) * 64;
    v8f acc[4] = {v8f{}, v8f{}, v8f{}, v8f{}};
#pragma unroll
    for (int k0 = 0; k0 < 64; k0 += 4) {
      v2f a;
      a.x = arow[k0 + khalf];
      a.y = arow[k0 + khalf + 1];
      const float* w0 = sW + (k0 + khalf) * 64;
      const float* w1 = sW + (k0 + khalf + 1) * 64;
#pragma unroll
      for (int n0 = 0; n0 < 4; ++n0) {
        v2f b;
        b.x = w0[n0 * 16 + lrow];
        b.y = w1[n0 * 16 + lrow];
        acc[n0] = __builtin_amdgcn_wmma_f32_16x16x4_f32(
            false, a, false, b, (short)0, acc[n0], false, false);
      }
    }
    const int moff = (lane >> 4) << 3;          // 0 or 8
#pragma unroll
    for (int n0 = 0; n0 < 4; ++n0) {
      const int col = n0 * 16 + lrow;
      const float bb = bias[col];
#pragma unroll
      for (int i = 0; i < 8; ++i) {
        const int r = row0 + moff + i;
        float v = acc[n0][i] + bb;
        if (res) v += res[(size_t)r * 64 + col];
        out[(size_t)r * 64 + col] = v;
      }
    }
  }
}

// -------------------------------------------------------------------------
// Row layernorm (D=64) with optional relu. One wave32 per row, __shfl_xor
// reduction (wave32-correct: width explicitly 32).
__global__ __launch_bounds__(256) void k_ln(
    const float* __restrict__ in, const float* __restrict__ g,
    const float* __restrict__ b, float* __restrict__ out, int rows, int relu) {
  const int lane = threadIdx.x & 31;
  const int wid  = (blockIdx.x * blockDim.x + threadIdx.x) >> 5;
  const int nw   = (gridDim.x * blockDim.x) >> 5;
  const float2 gg = ((const float2*)g)[lane];
  const float2 bb = ((const float2*)b)[lane];
  for (int r = wid; r < rows; r += nw) {
    float2 v = ((const float2*)(in + (size_t)r * Dd))[lane];
    float s  = v.x + v.y;
    float sq = v.x * v.x + v.y * v.y;
#pragma unroll
    for (int off = 16; off > 0; off >>= 1) {
      s  += __shfl_xor(s, off, 32);
      sq += __shfl_xor(sq, off, 32);
    }
    float mu  = s * (1.f / 64.f);
    float var = sq * (1.f / 64.f) - mu * mu;
    float rs  = rsqrtf(var + LN_EPS);
    float o0 = (v.x - mu) * rs * gg.x + bb.x;
    float o1 = (v.y - mu) * rs * gg.y + bb.y;
    if (relu) { o0 = fmaxf(o0, 0.f); o1 = fmaxf(o1, 0.f); }
    float2 o; o.x = o0; o.y = o1;
    ((float2*)(out + (size_t)r * Dd))[lane] = o;
  }
}

// Global add pool: vt[batch[n]] += h2[n]  (vt pre-seeded with vn)
__global__ __launch_bounds__(256) void k_pool(
    const float* __restrict__ h2, const int* __restrict__ batch,
    float* __restrict__ vt) {
  int i = blockIdx.x * blockDim.x + threadIdx.x;
  const int tot = Nn * 32;
  for (; i < tot; i += gridDim.x * blockDim.x) {
    int n = i >> 5, lane = i & 31;
    int gdx = batch[n];
    float2 v = ((const float2*)(h2 + (size_t)n * Dd))[lane];
    atomicAdd(&vt[(size_t)gdx * Dd + 2 * lane],     v.x);
    atomicAdd(&vt[(size_t)gdx * Dd + 2 * lane + 1], v.y);
  }
}

// h2[n] += vn[batch[n]]   (in place)
__global__ __launch_bounds__(256) void k_add_vn(
    float* __restrict__ h2, const int* __restrict__ batch,
    const float* __restrict__ vn) {
  int i = blockIdx.x * blockDim.x + threadIdx.x;
  const int tot = Nn * 32;
  for (; i < tot; i += gridDim.x * blockDim.x) {
    int n = i >> 5, lane = i & 31;
    int gdx = batch[n];
    float2 v = ((const float2*)(vn + (size_t)gdx * Dd))[lane];
    float2* p = (float2*)(h2 + (size_t)n * Dd) + lane;
    float2 h = *p;
    h.x += v.x; h.y += v.y;
    *p = h;
  }
}

// -------------------------------------------------------------------------
extern "C" void kernel_launch(void* const* d_in, const int* in_sizes, int n_in,
                              void* d_out, int out_size, void* d_ws, size_t ws_size,
                              hipStream_t stream) {
  (void)in_sizes; (void)n_in; (void)out_size; (void)ws_size;
  const int*   x        = (const int*)d_in[0];
  const int*   ei       = (const int*)d_in[1];
  const int*   eattr    = (const int*)d_in[2];
  const int*   batch    = (const int*)d_in[3];
  const float* atom_emb = (const float*)d_in[4];
  const float* bond_emb = (const float*)d_in[5];
  const float* gcn_W    = (const float*)d_in[6];
  const float* gcn_b    = (const float*)d_in[7];
  const float* norm_g   = (const float*)d_in[8];
  const float* norm_b   = (const float*)d_in[9];
  const float* vn_emb   = (const float*)d_in[10];
  const float* vn_W1    = (const float*)d_in[11];
  const float* vn_b1    = (const float*)d_in[12];
  const float* vn_g1    = (const float*)d_in[13];
  const float* vn_be1   = (const float*)d_in[14];
  const float* vn_W2    = (const float*)d_in[15];
  const float* vn_b2    = (const float*)d_in[16];
  const float* vn_g2    = (const float*)d_in[17];
  const float* vn_be2   = (const float*)d_in[18];
  float* out = (float*)d_out;

  const int* srcp = ei;
  const int* dstp = ei + Ee;

  // Workspace: 5 N x D f32 buffers + 3 G x D buffers  (~64 MB total)
  char* ws = (char*)d_ws;
  const size_t nd = (size_t)Nn * Dd * sizeof(float);
  float* bufA   = (float*)(ws);            // h (current)
  float* bufB   = (float*)(ws + 1 * nd);   // h2 / gemm out (ping-pong)
  float* bufMAX = (float*)(ws + 2 * nd);
  float* bufNUM = (float*)(ws + 3 * nd);
  float* bufDEN = (float*)(ws + 4 * nd);
  float* bufVN  = (float*)(ws + 5 * nd);
  float* bufVT  = bufVN + Gg * Dd;
  float* bufVT2 = bufVT + Gg * Dd;

  const dim3 blk(256);
  const int nbEdge = 2048;                       // grid-stride, wave per edge
  const int nbNode = 2048;                       // wave per node / elementwise
  const int nbGemmN = (Nn / 16 + 7) / 8;         // 391 blocks
  const int nbGemmG = (Gg / 16 + 7) / 8;         // 2 blocks

  auto genconv = [&](const float* W, const float* bias, const float* hx,
                     const float* res, float* outp) {
    (void)hipMemsetAsync(bufMAX, 0, nd, stream);
    (void)hipMemsetAsync(bufNUM, 0, nd, stream);
    (void)hipMemsetAsync(bufDEN, 0, nd, stream);
    k_msg_max<<<nbEdge, blk, 0, stream>>>(hx, srcp, dstp, eattr, bond_emb, bufMAX);
    k_msg_sum<<<nbEdge, blk, 0, stream>>>(hx, srcp, dstp, eattr, bond_emb,
                                          bufMAX, bufNUM, bufDEN);
    k_finalize<<<nbNode, blk, 0, stream>>>(hx, bufDEN, bufNUM);
    k_gemm64<<<nbGemmN, blk, 0, stream>>>(bufNUM, W, bias, res, outp, Nn);
  };

  // ---- Encoders + initial virtual node ----
  k_init_vn<<<(Gg * Dd + 255) / 256, blk, 0, stream>>>(vn_emb, bufVN);
  k_atom_encode<<<nbNode, blk, 0, stream>>>(x, atom_emb, vn_emb, bufA);

  // ---- Layer 0 (no residual) ----
  genconv(gcn_W, gcn_b, bufA, nullptr, bufB);
  std::swap(bufA, bufB);

  // ---- Layers 1..3 (res+ blocks with virtual-node update) ----
  for (int l = 1; l < Ll; ++l) {
    const int j = l - 1;
    k_ln<<<nbNode, blk, 0, stream>>>(bufA, norm_g + j * Dd, norm_b + j * Dd,
                                     bufB, Nn, 1);                    // h2
    (void)hipMemcpyAsync(bufVT, bufVN, (size_t)Gg * Dd * sizeof(float),
                         hipMemcpyDeviceToDevice, stream);             // vt = vn
    k_pool<<<nbNode, blk, 0, stream>>>(bufB, batch, bufVT);            // + pool(h2)
    k_gemm64<<<nbGemmG, blk, 0, stream>>>(bufVT, vn_W1 + j * Dd * Dd,
                                          vn_b1 + j * Dd, nullptr, bufVT2, Gg);
    k_ln<<<Gg / 8, blk, 0, stream>>>(bufVT2, vn_g1 + j * Dd, vn_be1 + j * Dd,
                                     bufVT, Gg, 1);
    k_gemm64<<<nbGemmG, blk, 0, stream>>>(bufVT, vn_W2 + j * Dd * Dd,
                                          vn_b2 + j * Dd, nullptr, bufVT2, Gg);
    k_ln<<<Gg / 8, blk, 0, stream>>>(bufVT2, vn_g2 + j * Dd, vn_be2 + j * Dd,
                                     bufVN, Gg, 1);                    // new vn
    k_add_vn<<<nbNode, blk, 0, stream>>>(bufB, batch, bufVN);          // h2 += vn[batch]
    genconv(gcn_W + l * Dd * Dd, gcn_b + l * Dd, bufB, bufA, bufB);    // + residual h
    std::swap(bufA, bufB);
  }

  // ---- Final layernorm ----
  k_ln<<<nbNode, blk, 0, stream>>>(bufA, norm_g + (Ll - 1) * Dd,
                                   norm_b + (Ll - 1) * Dd, out, Nn, 0);
}